// BertWithAdaThresholdLocContextPooling_46935402611426
// MI455X (gfx1250) — compile-verified
//
#include <hip/hip_runtime.h>
#include <hip/hip_bf16.h>

typedef __attribute__((ext_vector_type(2))) float v2f;
typedef __attribute__((ext_vector_type(8))) float v8f;

#define BB   32
#define HH   12
#define CC   512
#define DD   768
#define MM   8
#define EMB  768
#define BLK  8
#define NCLS 97
#define NER  6
#define OFFS 1
#define K1   (2*DD + NER)   // 1542
#define KP1  1600           // padded to multiple of 64
#define KBIL (EMB*BLK)      // 6144

// ---------------------------------------------------------------------------
// Kernel 1: per-batch prep.  hs = logsumexp over head mentions;
// ht_att[c] = normalize_c( sum_h (sum_m att[h,p0m,c])(sum_m att[h,p1m,c]) )
// ---------------------------------------------------------------------------
__global__ __launch_bounds__(256) void prep_kernel(
    const float* __restrict__ seq, const float* __restrict__ att,
    const int* __restrict__ epos, float* __restrict__ ws_ht,
    float* __restrict__ ws_hs)
{
  const int b = blockIdx.x;
  const int tid = threadIdx.x;
  __shared__ int p0[MM], p1[MM];
  __shared__ float red[256];

  if (tid < MM)            p0[tid]      = epos[(b*2 + 0)*MM + tid] + OFFS;
  else if (tid < 2*MM)     p1[tid - MM] = epos[(b*2 + 1)*MM + (tid - MM)] + OFFS;
  __syncthreads();

  // hs = logsumexp over the 8 head mentions, per feature d
  const float* sb = seq + (size_t)b * CC * DD;
  for (int d = tid; d < DD; d += 256) {
    float v[MM], mx = -1e30f;
    #pragma unroll
    for (int m = 0; m < MM; ++m) { v[m] = sb[(size_t)p0[m]*DD + d]; mx = fmaxf(mx, v[m]); }
    float s = 0.f;
    #pragma unroll
    for (int m = 0; m < MM; ++m) s += __expf(v[m] - mx);
    ws_hs[b*DD + d] = __logf(s) + mx;
  }

  // attention gather + head product, threads over context position c
  float lsum = 0.f;
  const float scale = 1.0f / (float)(HH * MM * MM);
  for (int c = tid; c < CC; c += 256) {
    float acc = 0.f;
    for (int h = 0; h < HH; ++h) {
      const float* ab = att + ((size_t)b*HH + h) * CC * CC;
      float s0 = 0.f, s1 = 0.f;
      #pragma unroll
      for (int m = 0; m < MM; ++m) {
        s0 += ab[(size_t)p0[m]*CC + c];
        s1 += ab[(size_t)p1[m]*CC + c];
      }
      acc = fmaf(s0, s1, acc);
    }
    acc *= scale;
    ws_ht[b*CC + c] = acc;
    lsum += acc;
  }

  // block reduction of lsum
  red[tid] = lsum;
  __syncthreads();
  for (int s = 128; s > 0; s >>= 1) {
    if (tid < s) red[tid] += red[tid + s];
    __syncthreads();
  }
  const float inv = 1.0f / (red[0] + 1e-5f);
  for (int c = tid; c < CC; c += 256) ws_ht[b*CC + c] *= inv;
}

// ---------------------------------------------------------------------------
// Kernel 2: rs[b,d] = sum_c seq[b,c,d] * ht[b,c]   (the 50 MB streaming read)
// ---------------------------------------------------------------------------
__global__ __launch_bounds__(256) void rs_kernel(
    const float* __restrict__ seq, const float* __restrict__ ws_ht,
    float* __restrict__ ws_rs)
{
  const int b = blockIdx.x;
  const int d = blockIdx.y * 256 + threadIdx.x;
  __shared__ float ht[CC];
  for (int c = threadIdx.x; c < CC; c += 256) ht[c] = ws_ht[b*CC + c];
  __syncthreads();
  const float* sp = seq + (size_t)b * CC * DD + d;
  float acc = 0.f;
  #pragma unroll 8
  for (int c = 0; c < CC; ++c) acc = fmaf(sp[(size_t)c * DD], ht[c], acc);
  ws_rs[b*DD + d] = acc;
}

// ---------------------------------------------------------------------------
// Kernel 3: build padded GEMM inputs X = [hs | rs | ner | 0-pad] (32 x 1600)
// ---------------------------------------------------------------------------
__global__ __launch_bounds__(256) void build_x_kernel(
    const float* __restrict__ hs, const float* __restrict__ rs,
    const float* __restrict__ ner, float* __restrict__ Xh,
    float* __restrict__ Xt)
{
  const int b = blockIdx.x;
  for (int k = threadIdx.x; k < KP1; k += 256) {
    float vh, vt;
    if (k < DD)            { float v = hs[b*DD + k];      vh = v; vt = v; }
    else if (k < 2*DD)     { float v = rs[b*DD + k - DD]; vh = v; vt = v; }
    else if (k < K1)       { int j = k - 2*DD;
                             vh = ner[(b*2 + 0)*NER + j];
                             vt = ner[(b*2 + 1)*NER + j]; }
    else                   { vh = 0.f; vt = 0.f; }
    Xh[b*KP1 + k] = vh;
    Xt[b*KP1 + k] = vt;
  }
}

// ---------------------------------------------------------------------------
// WMMA fp32 GEMM: out(32 x nreal) = [tanh]( X(32 x kp) * W^T + bias )
// W stored row-major (nreal x kreal).  One workgroup = 32x64 output tile;
// 8 waves -> (mi in 0..1) x (ni in 0..3) 16x16 tiles; K chunked by 64 via LDS.
// Fragment layout per CDNA5 ISA 7.12.2 (f32 A 16x4 / B 4x16 / C 16x16).
// ---------------------------------------------------------------------------
template <bool TANH>
__global__ __launch_bounds__(256) void gemm_wmma(
    const float* __restrict__ X, int ldx,
    const float* __restrict__ W, int ldw, int kreal, int kp,
    const float* __restrict__ bias,
    float* __restrict__ out, int ldo, int nreal)
{
  __shared__ float As[32][68];   // +4 pad: 16 rows hit 16 distinct banks
  __shared__ float Bs[64][65];   // +1 pad
  const int tid  = threadIdx.x;
  const int nbase = blockIdx.x * 64;
  const int wave = tid >> 5;
  const int lane = tid & 31;
  const int lo   = lane & 15;    // M row (A) / N col (B,C)
  const int hi   = lane >> 4;    // K half-select (A,B) / M+8 select (C)
  const int mi   = wave & 1;
  const int ni   = wave >> 1;

  v8f acc = {};
  for (int kb = 0; kb < kp; kb += 64) {
    for (int idx = tid; idx < 32*64; idx += 256) {
      int r = idx >> 6, k = idx & 63;
      As[r][k] = X[(size_t)r * ldx + kb + k];          // kb+k < kp == ldx
    }
    for (int idx = tid; idx < 64*64; idx += 256) {
      int nl = idx >> 6, k = idx & 63;
      int gn = nbase + nl, gk = kb + k;
      Bs[k][nl] = (gn < nreal && gk < kreal) ? W[(size_t)gn * ldw + gk] : 0.f;
    }
    __syncthreads();
    #pragma unroll
    for (int kk = 0; kk < 64; kk += 4) {
      v2f a, bf;
      a.x  = As[mi*16 + lo][kk + 2*hi];       // lanes 0-15: K=0,1 ; 16-31: K=2,3
      a.y  = As[mi*16 + lo][kk + 2*hi + 1];
      bf.x = Bs[kk + 2*hi][ni*16 + lo];
      bf.y = Bs[kk + 2*hi + 1][ni*16 + lo];
      acc = __builtin_amdgcn_wmma_f32_16x16x4_f32(
          false, a, false, bf, (short)0, acc, false, false);
    }
    __syncthreads();
  }

  const int col = nbase + ni*16 + lo;
  if (col < nreal) {
    const float bv = bias[col];
    #pragma unroll
    for (int r = 0; r < 8; ++r) {
      const int row = mi*16 + hi*8 + r;       // C layout: VGPR r -> M=r / M=r+8
      float v = acc[r] + bv;
      out[(size_t)row * ldo + col] = TANH ? tanhf(v) : v;
    }
  }
}

// ---------------------------------------------------------------------------
// Kernel 5: bl[b, g*64 + i*8 + j] = hs_f[b, g*8+i] * ts_f[b, g*8+j]
// ---------------------------------------------------------------------------
__global__ __launch_bounds__(256) void bl_kernel(
    const float* __restrict__ hsf, const float* __restrict__ tsf,
    float* __restrict__ bl)
{
  const int b = blockIdx.x;
  __shared__ float h[EMB], t[EMB];
  for (int i = threadIdx.x; i < EMB; i += 256) {
    h[i] = hsf[b*EMB + i];
    t[i] = tsf[b*EMB + i];
  }
  __syncthreads();
  for (int idx = threadIdx.x; idx < KBIL; idx += 256) {
    int g = idx >> 6, rem = idx & 63, i = rem >> 3, j = rem & 7;
    bl[(size_t)b * KBIL + idx] = h[g*BLK + i] * t[g*BLK + j];
  }
}

// ---------------------------------------------------------------------------
extern "C" void kernel_launch(void* const* d_in, const int* in_sizes, int n_in,
                              void* d_out, int out_size, void* d_ws, size_t ws_size,
                              hipStream_t stream) {
  const float* seq    = (const float*)d_in[0];   // (B,C,D)
  const float* att    = (const float*)d_in[1];   // (B,H,C,C)
  const float* ner    = (const float*)d_in[2];   // (B,2,NER)
  const float* W_head = (const float*)d_in[3];   // (EMB, K1)
  const float* b_head = (const float*)d_in[4];
  const float* W_tail = (const float*)d_in[5];
  const float* b_tail = (const float*)d_in[6];
  const float* W_bil  = (const float*)d_in[7];   // (NCLS, KBIL)
  const float* b_bil  = (const float*)d_in[8];
  const int*   epos   = (const int*)d_in[9];     // (B,2,M)
  float* logits = (float*)d_out;                 // (B, NCLS)

  float* w = (float*)d_ws;
  float* ws_ht  = w; w += BB*CC;
  float* ws_hs  = w; w += BB*DD;
  float* ws_rs  = w; w += BB*DD;
  float* ws_Xh  = w; w += BB*KP1;
  float* ws_Xt  = w; w += BB*KP1;
  float* ws_hsf = w; w += BB*EMB;
  float* ws_tsf = w; w += BB*EMB;
  float* ws_bl  = w; w += (size_t)BB*KBIL;

  prep_kernel<<<BB, 256, 0, stream>>>(seq, att, epos, ws_ht, ws_hs);
  rs_kernel<<<dim3(BB, DD/256), 256, 0, stream>>>(seq, ws_ht, ws_rs);
  build_x_kernel<<<BB, 256, 0, stream>>>(ws_hs, ws_rs, ner, ws_Xh, ws_Xt);

  gemm_wmma<true><<<EMB/64, 256, 0, stream>>>(ws_Xh, KP1, W_head, K1, K1, KP1,
                                              b_head, ws_hsf, EMB, EMB);
  gemm_wmma<true><<<EMB/64, 256, 0, stream>>>(ws_Xt, KP1, W_tail, K1, K1, KP1,
                                              b_tail, ws_tsf, EMB, EMB);
  bl_kernel<<<BB, 256, 0, stream>>>(ws_hsf, ws_tsf, ws_bl);
  gemm_wmma<false><<<2, 256, 0, stream>>>(ws_bl, KBIL, W_bil, KBIL, KBIL, KBIL,
                                          b_bil, logits, NCLS, NCLS);
}